// ChamferLoss_87170656240054
// MI455X (gfx1250) — compile-verified
//
#include <hip/hip_runtime.h>
#include <math.h>

typedef float v2f __attribute__((ext_vector_type(2)));
typedef float v8f __attribute__((ext_vector_type(8)));

#define BATCH   2
#define NPTS    8192
#define NT      (NPTS / 16)      // 512 tiles per dimension
#define WAVES   8                // waves per workgroup (256 threads, wave32)
#define STRIPS  (NT / WAVES)     // 64 row-strips per batch
#define JCHUNK  128              // tiles of J per block (2048 columns)
#define JCHUNKS (NT / JCHUNK)    // 4
#define FINF    0x7F800000u

__global__ void chamfer_init_kernel(unsigned* __restrict__ p, int n) {
    int i = blockIdx.x * blockDim.x + threadIdx.x;
    if (i < n) p[i] = FINF;
}

// One wave owns a 16-row strip of the distance matrix for a 2048-column chunk.
// Cross term via v_wmma_f32_16x16x4_f32. All min-reductions run in the raw
// squared-distance domain; since max(.,0) distributes over min, the clamp is
// applied once per reduced value right before each atomic (which also makes
// the uint-bit atomicMin ordering valid). sqrt is monotone -> applied once at
// the very end.
__global__ __launch_bounds__(256) void chamfer_tiles_kernel(
    const float* __restrict__ x1,   // (B,3,N): x points
    const float* __restrict__ y1,   // (B,3,N): y points
    unsigned* __restrict__ min1g,   // per x point j: min over y (B*N uints)
    unsigned* __restrict__ min2g)   // per y point i: min over x (B*N uints)
{
    __shared__ unsigned ldsCol[JCHUNK * 16];   // 8 KB column-min scratch

    const int bid   = blockIdx.x;
    const int b     = bid / (STRIPS * JCHUNKS);
    const int rem   = bid % (STRIPS * JCHUNKS);
    const int strip = rem / JCHUNKS;
    const int jc    = rem % JCHUNKS;

    const int tid  = threadIdx.x;
    const int wave = tid >> 5;
    const int lane = tid & 31;
    const bool hi  = lane >= 16;
    const int lane16 = lane & 15;

    for (int i = tid; i < JCHUNK * 16; i += 256) ldsCol[i] = FINF;
    __syncthreads();

    const float* X = x1 + (size_t)b * 3 * NPTS;
    const float* Y = y1 + (size_t)b * 3 * NPTS;

    // ---- Build A (16x4 f32): row m = (-2*y0, -2*y1, -2*y2, |y|^2) ----
    const int rowBase = (strip * WAVES + wave) * 16;
    const int row = rowBase + lane16;
    const float yx = Y[row];
    const float yy = Y[NPTS + row];
    const float yz = Y[2 * NPTS + row];
    const float ysq = yx * yx + yy * yy + yz * yz;
    v2f a;
    a.x = hi ? (-2.0f * yz) : (-2.0f * yx);   // K=2 : K=0
    a.y = hi ? ysq          : (-2.0f * yy);   // K=3 : K=1

    const float INF = __uint_as_float(FINF);
    float racc[8];
#pragma unroll
    for (int r = 0; r < 8; ++r) racc[r] = INF;

    const int jt0 = jc * JCHUNK;
#pragma unroll 2
    for (int jt = 0; jt < JCHUNK; ++jt) {
        const int col = (jt0 + jt) * 16 + lane16;
        // ---- Build B (4x16 f32): col j = (x0, x1, x2, 1) ----
        const float xx = X[col];
        const float xy = X[NPTS + col];
        const float xz = X[2 * NPTS + col];
        const float xsq = xx * xx + xy * xy + xz * xz;
        v2f bm;
        bm.x = hi ? xz   : xx;    // K=2 : K=0
        bm.y = hi ? 1.0f : xy;    // K=3 : K=1
        v8f c;
#pragma unroll
        for (int r = 0; r < 8; ++r) c[r] = xsq;   // C[i][j] = |x_j|^2

        // D[i][j] = |y_i|^2 + |x_j|^2 - 2*y_i.x_j  (unclamped d^2)
        v8f d = __builtin_amdgcn_wmma_f32_16x16x4_f32(
            false, a, false, bm, (short)0, c, false, false);

        float cm = INF;
#pragma unroll
        for (int r = 0; r < 8; ++r) {
            racc[r] = fminf(racc[r], d[r]);   // running row min (over j)
            cm = fminf(cm, d[r]);             // column min over this lane's 8 rows
        }
        // combine the two row-halves (same column index), clamp, publish
        cm = fminf(cm, __shfl_xor(cm, 16, 32));
        cm = fmaxf(cm, 0.0f);                 // clamp once, after the min
        if (!hi)
            atomicMin(&ldsCol[jt * 16 + lane16], __float_as_uint(cm));
    }

    // ---- Row mins: reduce across the 16 lanes of each half ----
#pragma unroll
    for (int r = 0; r < 8; ++r) {
        float v = racc[r];
        v = fminf(v, __shfl_xor(v, 1, 32));
        v = fminf(v, __shfl_xor(v, 2, 32));
        v = fminf(v, __shfl_xor(v, 4, 32));
        v = fminf(v, __shfl_xor(v, 8, 32));
        v = fmaxf(v, 0.0f);                   // clamp once, after the min
        if (lane16 == 0) {
            int rowIdx = rowBase + (hi ? 8 : 0) + r;
            atomicMin(&min2g[b * NPTS + rowIdx], __float_as_uint(v));
        }
    }

    // ---- Flush column mins (one global atomic per column per block) ----
    __syncthreads();
    for (int i = tid; i < JCHUNK * 16; i += 256)
        atomicMin(&min1g[b * NPTS + jt0 * 16 + i], ldsCol[i]);
}

// Sum sqrt(EPS + d2min) over all 2*B*N entries; emit mean(min1)+mean(min2).
__global__ __launch_bounds__(256) void chamfer_reduce_kernel(
    const unsigned* __restrict__ mins, float* __restrict__ out)
{
    __shared__ float s[256];
    const int tid = threadIdx.x;
    float sum = 0.0f;
    const int total = 2 * BATCH * NPTS;
    for (int i = tid; i < total; i += 256)
        sum += sqrtf(__uint_as_float(mins[i]) + 1e-8f);
    s[tid] = sum;
    __syncthreads();
    for (int off = 128; off > 0; off >>= 1) {
        if (tid < off) s[tid] += s[tid + off];
        __syncthreads();
    }
    if (tid == 0)
        out[0] = s[0] / (float)(BATCH * NPTS);   // sum1/BN + sum2/BN
}

extern "C" void kernel_launch(void* const* d_in, const int* in_sizes, int n_in,
                              void* d_out, int out_size, void* d_ws, size_t ws_size,
                              hipStream_t stream) {
    const float* x1 = (const float*)d_in[0];
    const float* y1 = (const float*)d_in[1];
    unsigned* min1 = (unsigned*)d_ws;                 // B*N uints
    unsigned* min2 = min1 + BATCH * NPTS;             // B*N uints
    const int nmins = 2 * BATCH * NPTS;               // 32768 (128 KB)

    chamfer_init_kernel<<<(nmins + 255) / 256, 256, 0, stream>>>(min1, nmins);
    chamfer_tiles_kernel<<<BATCH * STRIPS * JCHUNKS, 256, 0, stream>>>(
        x1, y1, min1, min2);
    chamfer_reduce_kernel<<<1, 256, 0, stream>>>(min1, (float*)d_out);
}